// NTS_model_22222160789556
// MI455X (gfx1250) — compile-verified
//
#include <hip/hip_runtime.h>
#include <hip/hip_bf16.h>

typedef __attribute__((ext_vector_type(2))) float v2f;
typedef __attribute__((ext_vector_type(8))) float v8f;

#define GF_RELU   1
#define GF_BIAS   2
#define GF_TRANSB 4

// ---------------------------------------------------------------------------
// K-loop helper for the fp32 WMMA GEMM. TRANSB selects the B addressing
// pattern; SCALED applies 0/1 edge masks (only needed for boundary tiles).
// ---------------------------------------------------------------------------
template <bool TRANSB, bool SCALED>
__device__ inline v8f wmma_kloop(const float* __restrict__ pA,
                                 const float* __restrict__ pB0,
                                 const float* __restrict__ pB1,
                                 size_t stepB, int K,
                                 float ascale, float bscale)
{
    v8f acc = {};
    for (int k0 = 0; k0 < K; k0 += 4) {
        v2f a = *(const v2f*)pA;  pA += 4;
        v2f b;
        if (TRANSB) {
            b = *(const v2f*)pB0;  pB0 += 4;
        } else {
            b.x = *pB0;  pB0 += stepB;
            b.y = *pB1;  pB1 += stepB;
        }
        if (SCALED) {
            a *= ascale;
            b *= bscale;
        }
        acc = __builtin_amdgcn_wmma_f32_16x16x4_f32(
            false, a, false, b, (short)0, acc, false, false);
    }
    return acc;
}

// ---------------------------------------------------------------------------
// Generic fp32 GEMM using V_WMMA_F32_16X16X4_F32. One 16x16 C tile per wave32.
// blockDim = (32,4): 4 waves, each a different M-tile. K must be a multiple
// of 4 (true for all call sites: 100 / 128 / 512 / 1000).
// Interior tiles (wave-uniform test) take an unscaled fast loop; boundary
// tiles clamp OOB lanes to a valid row/col and scale the fragment by 0, so
// EXEC stays all-ones through every WMMA.
// A: [M,K] lda, B: [K,N] ldb (or [N,K] if TRANSB), C: [M,N] ldc.
// ---------------------------------------------------------------------------
template <int FLAGS>
__global__ __launch_bounds__(128) void gemm_f32_wmma(
    const float* __restrict__ A, const float* __restrict__ B,
    const float* __restrict__ bias, float* __restrict__ C,
    int M, int N, int K, int lda, int ldb, int ldc)
{
    const int tileN = blockIdx.x;
    const int tileM = blockIdx.y * 4 + threadIdx.y;   // uniform per wave
    if (tileM * 16 >= M) return;                      // wave-uniform exit

    const int lane = threadIdx.x;      // 0..31
    const int half = lane >> 4;        // 0/1
    const int l    = lane & 15;
    const int am   = tileM * 16 + l;   // A row held by this lane
    const int bn   = tileN * 16 + l;   // B col held by this lane

    const bool interior = (tileM * 16 + 16 <= M) && (tileN * 16 + 16 <= N);

    const float ascale = (am < M) ? 1.f : 0.f;
    const float bscale = (bn < N) ? 1.f : 0.f;
    const int   amc    = (am < M) ? am : (M - 1);
    const int   bnc    = (bn < N) ? bn : (N - 1);

    const float* pA = A + (size_t)amc * lda + 2 * half;   // float2 per iter
    const float* pB0;
    const float* pB1 = nullptr;
    size_t stepB = 0;
    if (FLAGS & GF_TRANSB) {
        pB0 = B + (size_t)bnc * ldb + 2 * half;
    } else {
        pB0 = B + (size_t)(2 * half) * ldb + bnc;
        pB1 = pB0 + ldb;
        stepB = (size_t)4 * ldb;
    }

    v8f acc;
    if (interior) {     // wave-uniform branch: EXEC full inside either path
        acc = wmma_kloop<(FLAGS & GF_TRANSB) != 0, false>(
            pA, pB0, pB1, stepB, K, 1.f, 1.f);
    } else {
        acc = wmma_kloop<(FLAGS & GF_TRANSB) != 0, true>(
            pA, pB0, pB1, stepB, K, ascale, bscale);
    }

    const float bv = (FLAGS & GF_BIAS) ? bias[bnc] : 0.f;
    if (interior) {
#pragma unroll
        for (int r = 0; r < 8; ++r) {
            const int cm = tileM * 16 + r + 8 * half;  // C VGPR r -> M=r+8*half
            float v = acc[r] + bv;
            if (FLAGS & GF_RELU) v = v > 0.f ? v : 0.f;
            C[(size_t)cm * ldc + bn] = v;
        }
    } else if (bn < N) {
#pragma unroll
        for (int r = 0; r < 8; ++r) {
            const int cm = tileM * 16 + r + 8 * half;
            if (cm < M) {
                float v = acc[r] + bv;
                if (FLAGS & GF_RELU) v = v > 0.f ? v : 0.f;
                C[(size_t)cm * ldc + bn] = v;
            }
        }
    }
}

// ---------------------------------------------------------------------------
// Small helper kernels (deterministic fixed-order reductions)
// ---------------------------------------------------------------------------
__global__ void k_rowsum(const float* __restrict__ m, float* __restrict__ kvec,
                         int D, int N) {
    int d = blockIdx.x * blockDim.x + threadIdx.x;
    if (d >= D) return;
    float s = 0.f;
    for (int n = 0; n < N; ++n) s += m[(size_t)d * N + n];
    kvec[d] = s;
}

__global__ void k_scale_hp(float* __restrict__ Hp, const float* __restrict__ kvec,
                           int D, int F) {
    int i = blockIdx.x * blockDim.x + threadIdx.x;
    if (i >= D * F) return;
    int d = i / F;
    Hp[i] = Hp[i] / fmaxf(kvec[d], 1.0f);
}

__global__ void k_rownorm(const float* __restrict__ X, float* __restrict__ Y,
                          int M, int F) {
    int r = blockIdx.x * blockDim.x + threadIdx.x;
    if (r >= M) return;
    float s = 0.f;
    for (int f = 0; f < F; ++f) { float v = X[(size_t)r * F + f]; s += v * v; }
    float nrm = fmaxf(sqrtf(s), 1e-12f);
    for (int f = 0; f < F; ++f) Y[(size_t)r * F + f] = X[(size_t)r * F + f] / nrm;
}

__global__ void k_heads(const float* __restrict__ Hp,
                        const float* __restrict__ Wmu, const float* __restrict__ bmu,
                        const float* __restrict__ Weta, const float* __restrict__ beta_b,
                        const float* __restrict__ Wgam, const float* __restrict__ bgam,
                        float* __restrict__ mu, float* __restrict__ eta,
                        float* __restrict__ gam, int D, int F) {
    int d = blockIdx.x * blockDim.x + threadIdx.x;
    if (d >= D) return;
    float smu = 0.f, seta = 0.f, sgam = 0.f;
    for (int f = 0; f < F; ++f) {
        float h = Hp[(size_t)d * F + f];
        smu  += h * Wmu[f];
        seta += h * Weta[f];
        sgam += h * Wgam[f];
    }
    mu[d]  = fmaxf(smu  + bmu[0],    0.f);
    eta[d] = fmaxf(seta + beta_b[0], 0.f);
    gam[d] = fmaxf(sgam + bgam[0],   0.f);
}

// top-(1-sparsity) mask of w_m with stable (argsort) tie-breaking
__global__ void k_prune(const int* __restrict__ epoch, const int* __restrict__ epochs,
                        const float* __restrict__ wm, const float* __restrict__ Wbeta,
                        float* __restrict__ wpr, int F) {
    int i = blockIdx.x * blockDim.x + threadIdx.x;
    if (i >= F) return;
    float sp = ((float)epoch[0] / (float)epochs[0]) * 0.3f;
    int j = (int)rintf(sp * (float)F);   // round-half-even == python round
    float wi = wm[i];
    int rank = 0;
    for (int l = 0; l < F; ++l) {
        float wl = wm[l];
        if (wl < wi || (wl == wi && l < i)) rank++;
    }
    wpr[i] = (rank >= j) ? Wbeta[i] : 0.f;
}

// y[i] = dot(A[i, :K], x)
__global__ void k_gemv(const float* __restrict__ A, const float* __restrict__ x,
                       float* __restrict__ y, int M, int K, int lda) {
    int r = blockIdx.x * blockDim.x + threadIdx.x;
    if (r >= M) return;
    float s = 0.f;
    for (int k = 0; k < K; ++k) s += A[(size_t)r * lda + k] * x[k];
    y[r] = s;
}

__global__ void k_normreduce(const float* __restrict__ y, float* __restrict__ nrm, int n) {
    __shared__ float sd[256];
    float s = 0.f;
    for (int i = threadIdx.x; i < n; i += 256) { float v = y[i]; s += v * v; }
    sd[threadIdx.x] = s;
    __syncthreads();
    for (int o = 128; o > 0; o >>= 1) {
        if (threadIdx.x < o) sd[threadIdx.x] += sd[threadIdx.x + o];
        __syncthreads();
    }
    if (threadIdx.x == 0) nrm[0] = fmaxf(sqrtf(sd[0]), 1e-12f);
}

__global__ void k_vscale(const float* __restrict__ y, const float* __restrict__ nrm,
                         float* __restrict__ out, int n) {
    int i = blockIdx.x * blockDim.x + threadIdx.x;
    if (i >= n) return;
    out[i] = y[i] / nrm[0];
}

__global__ void k_zd(const float* __restrict__ s_mat, const float* __restrict__ kvec,
                     float* __restrict__ Zd, int D, int F) {
    int d = blockIdx.x * blockDim.x + threadIdx.x;
    if (d >= D) return;
    float s = 0.f;
    for (int f = 0; f < F; ++f) { float v = s_mat[(size_t)d * F + f]; s += v * v; }
    Zd[d] = fmaxf(0.5f * (s - kvec[d]), 0.f);
}

__global__ void k_lambda(const float* __restrict__ mu, const float* __restrict__ betaD,
                         const float* __restrict__ eta, const float* __restrict__ gam,
                         const float* __restrict__ Zd, const float* __restrict__ himg,
                         const float* __restrict__ Wm2, const float* __restrict__ bm2,
                         float* __restrict__ out, int D, int HD) {
    int d = blockIdx.x * blockDim.x + threadIdx.x;
    if (d >= D) return;
    float lt = mu[d] + betaD[d] + eta[d] * expf(-gam[d] * Zd[d]);
    lt = 1.f / (1.f + expf(-lt));
    float li = bm2[0];
    for (int h = 0; h < HD; ++h) li += himg[(size_t)d * HD + h] * Wm2[h];
    out[d] = 1.f / (1.f + expf(-(lt + li)));
}

__global__ void k_E(const float* __restrict__ Zmat, float* __restrict__ E, int N) {
    int idx = blockIdx.x * blockDim.x + threadIdx.x;
    if (idx >= N * N) return;
    int i = idx / N, j = idx - i * N;
    float t = 2.f - 2.f * Zmat[idx];
    E[idx] = (i == j) ? 0.f : expf(-t * t);
}

// Z_event[d,i,j] = (i<k_d && j<k_d) ? E[i,j] : 0 ; one block per (d,i) row,
// 250 aligned float4 stores per row (N=1000). Pure streaming, store-bound;
// E (4MB) stays resident in L2 across the 100x replication.
__global__ __launch_bounds__(256) void k_zevent(const float* __restrict__ E,
                                                const float* __restrict__ kvec,
                                                float* __restrict__ Zev, int N) {
    int di = blockIdx.x;                 // d*N + i
    int d = di / N;
    int i = di - d * N;
    int kd = (int)(kvec[d] + 0.5f);
    int j4 = threadIdx.x * 4;
    if (j4 >= N) return;
    float4 out;
    if (i >= kd) {
        out.x = 0.f; out.y = 0.f; out.z = 0.f; out.w = 0.f;
    } else {
        const float4 e = *(const float4*)(E + (size_t)i * N + j4);
        out.x = (j4 + 0 < kd) ? e.x : 0.f;
        out.y = (j4 + 1 < kd) ? e.y : 0.f;
        out.z = (j4 + 2 < kd) ? e.z : 0.f;
        out.w = (j4 + 3 < kd) ? e.w : 0.f;
    }
    *(float4*)(Zev + (size_t)di * N + j4) = out;
}

// ---------------------------------------------------------------------------
static inline dim3 gemm_grid(int M, int N) {
    return dim3((unsigned)((N + 15) / 16), (unsigned)((M + 63) / 64));
}

extern "C" void kernel_launch(void* const* d_in, const int* in_sizes, int n_in,
                              void* d_out, int out_size, void* d_ws, size_t ws_size,
                              hipStream_t stream) {
    const int N = 1000, D = 100, F = 100, TI = 512, HD = 128;

    const int*   epoch  = (const int*)d_in[0];
    const int*   epochs = (const int*)d_in[1];
    const float* adj    = (const float*)d_in[2];    // [N,N]
    const float* masks  = (const float*)d_in[3];    // [D,N]
    const float* bows   = (const float*)d_in[4];    // [N,F]
    const float* imgf   = (const float*)d_in[5];    // [D,TI]
    const float* Wg1    = (const float*)d_in[6];
    const float* Wg2    = (const float*)d_in[7];
    const float* Wh1    = (const float*)d_in[8];
    const float* bh1    = (const float*)d_in[9];
    const float* Wh2    = (const float*)d_in[10];
    const float* bh2    = (const float*)d_in[11];
    const float* Wmu2   = (const float*)d_in[12];
    const float* bmu2   = (const float*)d_in[13];
    const float* Weta2  = (const float*)d_in[14];
    const float* beta2  = (const float*)d_in[15];
    const float* Wgam2  = (const float*)d_in[16];
    const float* bgam2  = (const float*)d_in[17];
    const float* wm     = (const float*)d_in[18];
    const float* Wbeta  = (const float*)d_in[19];
    const float* Wm1    = (const float*)d_in[20];
    const float* bm1    = (const float*)d_in[21];
    const float* Wm2    = (const float*)d_in[22];
    const float* bm2    = (const float*)d_in[23];

    // workspace layout (floats, 16-float aligned)
    float* ws    = (float*)d_ws;
    float* T0    = ws + 0;        // [N,F]
    float* H1b   = ws + 100000;   // [N,F]
    float* T1    = ws + 200000;   // [N,F]
    float* Hh1   = ws + 300000;   // [N,F]
    float* Heli  = ws + 400000;   // [N,F]
    float* Hn    = ws + 500000;   // [N,F]
    float* Hp    = ws + 600000;   // [D,F]
    float* kvec  = ws + 610000;   // [D]
    float* wpr   = ws + 610112;   // [F]
    float* ytmp  = ws + 610224;   // [N]
    float* nrm   = ws + 611232;   // [1]
    float* betaD = ws + 611248;   // [D]
    float* muv   = ws + 611360;   // [D]
    float* Zd    = ws + 611472;   // [D]
    float* smat  = ws + 611584;   // [D,F]
    float* himg  = ws + 621584;   // [D,HD]
    float* Ebuf  = ws + 634384;   // [N,N]

    // output layout: (lambda_total, Z_, beta_, gamma, eta, Z_event, H)
    float* out      = (float*)d_out;
    float* o_lambda = out;                 // [D]
    float* o_Z      = out + 100;           // [N,N]
    float* o_beta   = out + 1000100;       // [N]
    float* o_gamma  = out + 1001100;       // [D]
    float* o_eta    = out + 1001200;       // [D]
    float* o_Zev    = out + 1001300;       // [D,N,N]
    float* o_H      = out + 101001300;     // [N,F]

    const dim3 gb(32, 4);

    // GCN layer 1: T0 = bows@Wg1 ; H1 = relu(adj@T0)
    gemm_f32_wmma<0><<<gemm_grid(N, F), gb, 0, stream>>>(bows, Wg1, nullptr, T0,
        N, F, F, F, F, F);
    k_rowsum<<<1, 128, 0, stream>>>(masks, kvec, D, N);
    gemm_f32_wmma<GF_RELU><<<gemm_grid(N, F), gb, 0, stream>>>(adj, T0, nullptr, H1b,
        N, F, N, N, F, F);
    // GCN layer 2: T1 = H1@Wg2 ; H = adj@T1  (H straight into d_out)
    gemm_f32_wmma<0><<<gemm_grid(N, F), gb, 0, stream>>>(H1b, Wg2, nullptr, T1,
        N, F, F, F, F, F);
    gemm_f32_wmma<0><<<gemm_grid(N, F), gb, 0, stream>>>(adj, T1, nullptr, o_H,
        N, F, N, N, F, F);
    // doc pooling: Hp = (masks@H)/max(cnt,1)
    gemm_f32_wmma<0><<<gemm_grid(D, F), gb, 0, stream>>>(masks, o_H, nullptr, Hp,
        D, F, N, N, F, F);
    k_scale_hp<<<(D * F + 255) / 256, 256, 0, stream>>>(Hp, kvec, D, F);
    // H_eli = relu(relu(H@Wh1+b)@Wh2+b); then row-l2-normalize
    gemm_f32_wmma<GF_RELU | GF_BIAS><<<gemm_grid(N, F), gb, 0, stream>>>(
        o_H, Wh1, bh1, Hh1, N, F, F, F, F, F);
    gemm_f32_wmma<GF_RELU | GF_BIAS><<<gemm_grid(N, F), gb, 0, stream>>>(
        Hh1, Wh2, bh2, Heli, N, F, F, F, F, F);
    k_rownorm<<<(N + 127) / 128, 128, 0, stream>>>(Heli, Hn, N, F);
    // heads mu/eta/gamma (eta, gamma directly into d_out)
    k_heads<<<1, 128, 0, stream>>>(Hp, Wmu2, bmu2, Weta2, beta2, Wgam2, bgam2,
                                   muv, o_eta, o_gamma, D, F);
    // pruned beta path
    k_prune<<<1, 128, 0, stream>>>(epoch, epochs, wm, Wbeta, wpr, F);
    k_gemv<<<(N + 127) / 128, 128, 0, stream>>>(o_H, wpr, ytmp, N, F, F);
    k_normreduce<<<1, 256, 0, stream>>>(ytmp, nrm, N);
    k_vscale<<<(N + 127) / 128, 128, 0, stream>>>(ytmp, nrm, o_beta, N);
    k_gemv<<<1, 128, 0, stream>>>(masks, o_beta, betaD, D, N, N);
    // Z_ = relu(Hn @ Hn^T) directly into d_out
    gemm_f32_wmma<GF_RELU | GF_TRANSB><<<gemm_grid(N, N), gb, 0, stream>>>(
        Hn, Hn, nullptr, o_Z, N, N, F, F, F, N);
    // s = masks@Hn ; Z = relu(0.5*(||s||^2 - k))
    gemm_f32_wmma<0><<<gemm_grid(D, F), gb, 0, stream>>>(masks, Hn, nullptr, smat,
        D, F, N, N, F, F);
    k_zd<<<1, 128, 0, stream>>>(smat, kvec, Zd, D, F);
    // image branch + final lambda
    gemm_f32_wmma<GF_RELU | GF_BIAS><<<gemm_grid(D, HD), gb, 0, stream>>>(
        imgf, Wm1, bm1, himg, D, HD, TI, TI, HD, HD);
    k_lambda<<<1, 128, 0, stream>>>(muv, betaD, o_eta, o_gamma, Zd, himg,
                                    Wm2, bm2, o_lambda, D, HD);
    // E and the 400MB Z_event stream
    k_E<<<(N * N + 255) / 256, 256, 0, stream>>>(o_Z, Ebuf, N);
    k_zevent<<<D * N, 256, 0, stream>>>(Ebuf, kvec, o_Zev, N);
}